// GraphAttentionLayer_43430709297910
// MI455X (gfx1250) — compile-verified
//
#include <hip/hip_runtime.h>
#include <math.h>

typedef float v2f __attribute__((ext_vector_type(2)));
typedef float v8f __attribute__((ext_vector_type(8)));

#define Hdim 128
#define ETdim 8

// ---------------------------------------------------------------- init
__global__ __launch_bounds__(256) void init_kernel(float* agg, float* sm,
                                                   unsigned* mx, int N) {
  size_t i = (size_t)blockIdx.x * blockDim.x + threadIdx.x;
  size_t NH = (size_t)N * Hdim;
  if (i < NH) agg[i] = 0.0f;
  if (i < (size_t)N) { sm[i] = 0.0f; mx[i] = 0u; }
}

// ------------------------------------------- edge-type tables (tiny GEMM)
// EK = edge_emb @ Wk (no bias; bias lives in K0), EV = edge_emb @ Wv,
// ebias[t] = edge_emb[t] . Web + beb
__global__ __launch_bounds__(128) void prep_edge_tables_kernel(
    const float* __restrict__ edge_emb, const float* __restrict__ Wk,
    const float* __restrict__ Wv, const float* __restrict__ Web,
    const float* __restrict__ beb, float* __restrict__ EK,
    float* __restrict__ EV, float* __restrict__ ebias) {
  int t = blockIdx.x;          // edge type
  int col = threadIdx.x;       // output column
  float sk = 0.0f, sv = 0.0f;
  for (int k = 0; k < Hdim; ++k) {
    float e = edge_emb[t * Hdim + k];
    sk += e * Wk[k * Hdim + col];
    sv += e * Wv[k * Hdim + col];
  }
  EK[t * Hdim + col] = sk;
  EV[t * Hdim + col] = sv;
  if (col == 0) {
    float s = beb[0];
    for (int k = 0; k < Hdim; ++k) s += edge_emb[t * Hdim + k] * Web[k];
    ebias[t] = s;
  }
}

// ---------------------------------------------------- node projections (WMMA)
// One wave computes a 16-node x 128-col tile of hidden @ W + b.
// blockIdx.y selects (Wq,bq,Q) / (Wk,bk,K0) / (Wv,bv,V0).
__global__ __launch_bounds__(128) void node_proj_kernel(
    const float* __restrict__ hidden, const float* __restrict__ Wq,
    const float* __restrict__ Wk, const float* __restrict__ Wv,
    const float* __restrict__ bq, const float* __restrict__ bk,
    const float* __restrict__ bv, float* __restrict__ Q,
    float* __restrict__ K0, float* __restrict__ V0, int N, int ntiles) {
  int wave = threadIdx.x >> 5;
  int lane = threadIdx.x & 31;
  int tile = blockIdx.x * 4 + wave;
  if (tile >= ntiles) return;
  const float* W; const float* b; float* out;
  if (blockIdx.y == 0)      { W = Wq; b = bq; out = Q;  }
  else if (blockIdx.y == 1) { W = Wk; b = bk; out = K0; }
  else                      { W = Wv; b = bv; out = V0; }

  int row = lane & 15;        // M index for A, N index for B/C
  int khalf = lane >> 4;      // which K pair this lane holds
  int node = tile * 16 + row;
  if (node >= N) node = N - 1;  // clamp (N is a multiple of 16 here anyway)

  v8f acc[8] = {};
  const float* arow = hidden + (size_t)node * Hdim;
  for (int k0 = 0; k0 < Hdim; k0 += 4) {
    int ka = k0 + 2 * khalf;
    v2f a; a.x = arow[ka]; a.y = arow[ka + 1];
#pragma unroll
    for (int t = 0; t < 8; ++t) {
      int col = t * 16 + row;
      v2f bf; bf.x = W[ka * Hdim + col]; bf.y = W[(ka + 1) * Hdim + col];
      acc[t] = __builtin_amdgcn_wmma_f32_16x16x4_f32(
          false, a, false, bf, (short)0, acc[t], false, false);
    }
  }
#pragma unroll
  for (int t = 0; t < 8; ++t) {
    int col = t * 16 + row;
    float bb = b[col];
#pragma unroll
    for (int i = 0; i < 8; ++i) {
      int m = i + 8 * khalf;
      int r = tile * 16 + m;
      if (r < N) out[(size_t)r * Hdim + col] = acc[t][i] + bb;
    }
  }
}

// -------------------------------------------------------- edge logits + max
// One wave per edge; 4 floats/lane; shuffle reduce; ordered-uint atomicMax.
__device__ __forceinline__ unsigned float_to_ordered(float f) {
  unsigned u = __float_as_uint(f);
  return u ^ (unsigned)(((int)u >> 31) | 0x80000000);
}
__device__ __forceinline__ float ordered_to_float(unsigned u) {
  return __uint_as_float((u & 0x80000000u) ? (u ^ 0x80000000u) : ~u);
}

__global__ __launch_bounds__(256) void edge_logit_kernel(
    const int* __restrict__ edge_index, const int* __restrict__ edge_type,
    const float* __restrict__ Q, const float* __restrict__ K0,
    const float* __restrict__ EK, const float* __restrict__ ebias,
    float* __restrict__ logits, unsigned* __restrict__ mx, int E) {
  int wave = threadIdx.x >> 5;
  int lane = threadIdx.x & 31;
  int e = blockIdx.x * 8 + wave;
  if (e >= E) return;
  int src = edge_index[e];
  int tgt = edge_index[E + e];
  int t = edge_type[e];
  int c = lane * 4;
  float4 qv = *(const float4*)(Q + (size_t)tgt * Hdim + c);
  float4 kv = *(const float4*)(K0 + (size_t)src * Hdim + c);
  float4 ev = *(const float4*)(EK + t * Hdim + c);
  float p = qv.x * (kv.x + ev.x) + qv.y * (kv.y + ev.y) +
            qv.z * (kv.z + ev.z) + qv.w * (kv.w + ev.w);
#pragma unroll
  for (int m = 16; m > 0; m >>= 1) p += __shfl_xor(p, m, 32);
  if (lane == 0) {
    float l = p * 0.08838834764831845f + ebias[t];  // 1/sqrt(128)
    logits[e] = l;
    atomicMax(mx + tgt, float_to_ordered(l));
  }
}

// ------------------------------------------------------------- exp + segsum
__global__ __launch_bounds__(256) void edge_exp_kernel(
    const int* __restrict__ edge_index, const float* __restrict__ logits,
    const unsigned* __restrict__ mx, float* __restrict__ ex,
    float* __restrict__ sm, int E) {
  int e = blockIdx.x * 256 + threadIdx.x;
  if (e >= E) return;
  int tgt = edge_index[E + e];
  float m = ordered_to_float(mx[tgt]);
  float x = expf(logits[e] - m);
  ex[e] = x;
  atomicAdd(sm + tgt, x);
}

// --------------------------------------------------------- weighted scatter
__global__ __launch_bounds__(256) void edge_scatter_kernel(
    const int* __restrict__ edge_index, const int* __restrict__ edge_type,
    const float* __restrict__ V0, const float* __restrict__ EV,
    const float* __restrict__ ex, const float* __restrict__ sm,
    float* __restrict__ agg, int E) {
  int wave = threadIdx.x >> 5;
  int lane = threadIdx.x & 31;
  int e = blockIdx.x * 8 + wave;
  if (e >= E) return;
  int src = edge_index[e];
  int tgt = edge_index[E + e];
  int t = edge_type[e];
  float w = ex[e] / sm[tgt];
  int c = lane * 4;
  float4 vv = *(const float4*)(V0 + (size_t)src * Hdim + c);
  float4 ev = *(const float4*)(EV + t * Hdim + c);
  float* dst = agg + (size_t)tgt * Hdim + c;
  atomicAdd(dst + 0, (vv.x + ev.x) * w);
  atomicAdd(dst + 1, (vv.y + ev.y) * w);
  atomicAdd(dst + 2, (vv.z + ev.z) * w);
  atomicAdd(dst + 3, (vv.w + ev.w) * w);
}

// ------------------------------------------------- FFN layer 1 (WMMA, K=256)
// u1 = silu([hidden | agg] @ W1 + b1)
__global__ __launch_bounds__(128) void ffn1_kernel(
    const float* __restrict__ hidden, const float* __restrict__ agg,
    const float* __restrict__ W1, const float* __restrict__ b1,
    float* __restrict__ u1, int N, int ntiles) {
  int wave = threadIdx.x >> 5;
  int lane = threadIdx.x & 31;
  int tile = blockIdx.x * 4 + wave;
  if (tile >= ntiles) return;
  int row = lane & 15;
  int khalf = lane >> 4;
  int node = tile * 16 + row;
  if (node >= N) node = N - 1;

  v8f acc[8] = {};
  const float* hrow = hidden + (size_t)node * Hdim;
  const float* grow = agg + (size_t)node * Hdim;
  for (int k0 = 0; k0 < 2 * Hdim; k0 += 4) {
    int ka = k0 + 2 * khalf;
    const float* ap = (ka < Hdim) ? (hrow + ka) : (grow + (ka - Hdim));
    v2f a; a.x = ap[0]; a.y = ap[1];
#pragma unroll
    for (int t = 0; t < 8; ++t) {
      int col = t * 16 + row;
      v2f bf; bf.x = W1[ka * Hdim + col]; bf.y = W1[(ka + 1) * Hdim + col];
      acc[t] = __builtin_amdgcn_wmma_f32_16x16x4_f32(
          false, a, false, bf, (short)0, acc[t], false, false);
    }
  }
#pragma unroll
  for (int t = 0; t < 8; ++t) {
    int col = t * 16 + row;
    float bb = b1[col];
#pragma unroll
    for (int i = 0; i < 8; ++i) {
      int m = i + 8 * khalf;
      int r = tile * 16 + m;
      if (r < N) {
        float u = acc[t][i] + bb;
        u1[(size_t)r * Hdim + col] = u / (1.0f + expf(-u));  // silu
      }
    }
  }
}

// --------------------------------------- FFN layer 2 + residual (WMMA, K=128)
// xb = hidden + u1 @ W2 + b2
__global__ __launch_bounds__(128) void ffn2_kernel(
    const float* __restrict__ hidden, const float* __restrict__ u1,
    const float* __restrict__ W2, const float* __restrict__ b2,
    float* __restrict__ xb, int N, int ntiles) {
  int wave = threadIdx.x >> 5;
  int lane = threadIdx.x & 31;
  int tile = blockIdx.x * 4 + wave;
  if (tile >= ntiles) return;
  int row = lane & 15;
  int khalf = lane >> 4;
  int node = tile * 16 + row;
  if (node >= N) node = N - 1;

  v8f acc[8] = {};
  const float* arow = u1 + (size_t)node * Hdim;
  for (int k0 = 0; k0 < Hdim; k0 += 4) {
    int ka = k0 + 2 * khalf;
    v2f a; a.x = arow[ka]; a.y = arow[ka + 1];
#pragma unroll
    for (int t = 0; t < 8; ++t) {
      int col = t * 16 + row;
      v2f bf; bf.x = W2[ka * Hdim + col]; bf.y = W2[(ka + 1) * Hdim + col];
      acc[t] = __builtin_amdgcn_wmma_f32_16x16x4_f32(
          false, a, false, bf, (short)0, acc[t], false, false);
    }
  }
#pragma unroll
  for (int t = 0; t < 8; ++t) {
    int col = t * 16 + row;
    float bb = b2[col];
#pragma unroll
    for (int i = 0; i < 8; ++i) {
      int m = i + 8 * khalf;
      int r = tile * 16 + m;
      if (r < N)
        xb[(size_t)r * Hdim + col] =
            hidden[(size_t)r * Hdim + col] + acc[t][i] + bb;
    }
  }
}

// ------------------------------------------------------------- LayerNorm
__global__ __launch_bounds__(256) void ln_kernel(
    const float* __restrict__ xb, const float* __restrict__ gamma,
    const float* __restrict__ beta, float* __restrict__ out, int N) {
  int wave = threadIdx.x >> 5;
  int lane = threadIdx.x & 31;
  int node = blockIdx.x * 8 + wave;
  if (node >= N) return;
  int c = lane * 4;
  float4 xv = *(const float4*)(xb + (size_t)node * Hdim + c);
  float s = xv.x + xv.y + xv.z + xv.w;
#pragma unroll
  for (int m = 16; m > 0; m >>= 1) s += __shfl_xor(s, m, 32);
  float mu = s * (1.0f / Hdim);
  float dx = xv.x - mu, dy = xv.y - mu, dz = xv.z - mu, dw = xv.w - mu;
  float v = dx * dx + dy * dy + dz * dz + dw * dw;
#pragma unroll
  for (int m = 16; m > 0; m >>= 1) v += __shfl_xor(v, m, 32);
  float inv = rsqrtf(v * (1.0f / Hdim) + 1e-5f);
  float4 g = *(const float4*)(gamma + c);
  float4 bt = *(const float4*)(beta + c);
  float4 o;
  o.x = dx * inv * g.x + bt.x;
  o.y = dy * inv * g.y + bt.y;
  o.z = dz * inv * g.z + bt.z;
  o.w = dw * inv * g.w + bt.w;
  *(float4*)(out + (size_t)node * Hdim + c) = o;
}

// ================================================================ host
extern "C" void kernel_launch(void* const* d_in, const int* in_sizes, int n_in,
                              void* d_out, int out_size, void* d_ws,
                              size_t ws_size, hipStream_t stream) {
  const float* hidden = (const float*)d_in[0];
  const int* edge_index = (const int*)d_in[1];
  const int* edge_type = (const int*)d_in[2];
  const float* edge_emb = (const float*)d_in[3];
  const float* Wq = (const float*)d_in[4];
  const float* bq = (const float*)d_in[5];
  const float* Wk = (const float*)d_in[6];
  const float* bk = (const float*)d_in[7];
  const float* Wv = (const float*)d_in[8];
  const float* bv = (const float*)d_in[9];
  const float* Web = (const float*)d_in[10];
  const float* beb = (const float*)d_in[11];
  const float* W1 = (const float*)d_in[12];
  const float* b1 = (const float*)d_in[13];
  const float* W2 = (const float*)d_in[14];
  const float* b2 = (const float*)d_in[15];
  const float* gamma = (const float*)d_in[16];
  const float* beta = (const float*)d_in[17];
  float* out = (float*)d_out;

  const int N = in_sizes[0] / Hdim;
  const int E = in_sizes[2];
  const size_t NH = (size_t)N * Hdim;

  // workspace layout (floats)
  float* ws = (float*)d_ws;
  float* Q  = ws;
  float* K0 = ws + NH;
  float* V0 = ws + 2 * NH;
  float* agg = ws + 3 * NH;
  float* EK = ws + 4 * NH;
  float* EV = EK + ETdim * Hdim;
  float* ebias = EV + ETdim * Hdim;
  unsigned* mx = (unsigned*)(ebias + ETdim);
  float* sm = (float*)(mx + N);
  float* logits = sm + N;
  float* ex = logits + E;
  float* u1 = Q;   // reuse after edge phase
  float* xb = K0;  // reuse after edge phase

  const int ntiles = (N + 15) / 16;
  const int tblocks = (ntiles + 3) / 4;

  init_kernel<<<(int)((NH + 255) / 256), 256, 0, stream>>>(agg, sm, mx, N);
  prep_edge_tables_kernel<<<ETdim, 128, 0, stream>>>(edge_emb, Wk, Wv, Web,
                                                     beb, EK, EV, ebias);
  node_proj_kernel<<<dim3(tblocks, 3), 128, 0, stream>>>(
      hidden, Wq, Wk, Wv, bq, bk, bv, Q, K0, V0, N, ntiles);
  edge_logit_kernel<<<(E + 7) / 8, 256, 0, stream>>>(
      edge_index, edge_type, Q, K0, EK, ebias, logits, mx, E);
  edge_exp_kernel<<<(E + 255) / 256, 256, 0, stream>>>(edge_index, logits, mx,
                                                       ex, sm, E);
  edge_scatter_kernel<<<(E + 7) / 8, 256, 0, stream>>>(
      edge_index, edge_type, V0, EV, ex, sm, agg, E);
  ffn1_kernel<<<tblocks, 128, 0, stream>>>(hidden, agg, W1, b1, u1, N, ntiles);
  ffn2_kernel<<<tblocks, 128, 0, stream>>>(hidden, u1, W2, b2, xb, N, ntiles);
  ln_kernel<<<(N + 7) / 8, 256, 0, stream>>>(xb, gamma, beta, out, N);
}